// QueryAndGroup_46222438040237
// MI455X (gfx1250) — compile-verified
//
#include <hip/hip_runtime.h>

// Problem constants (from the reference)
#define BB      2
#define NN      16384
#define PP      4096
#define CC      64
#define NSAMPLE 32
#define RADIUS2 0.04f

#define WAVES_PER_BLOCK 8
#define THREADS (WAVES_PER_BLOCK * 32)

typedef __attribute__((ext_vector_type(2))) float v2f;
typedef __attribute__((ext_vector_type(8))) float v8f;

__global__ __launch_bounds__(THREADS)
void qg_wmma_kernel(const float* __restrict__ xyz,       // (B, N, 3)
                    const float* __restrict__ new_xyz,   // (B, P, 3)
                    const float* __restrict__ features,  // (B, C, N)
                    float* __restrict__ out)             // (B, 3+C, P, 32)
{
    // Per-wave private index buffer: 16 queries x 32 slots
    __shared__ int s_idx[WAVES_PER_BLOCK][16][NSAMPLE];

    const int lane  = threadIdx.x & 31;
    const int wave  = threadIdx.x >> 5;
    const int tile  = blockIdx.x * WAVES_PER_BLOCK + wave; // 16-query tile
    const int gq    = tile * 16;
    const int b     = gq / PP;
    const int pbase = gq % PP;          // P % 16 == 0, tiles never straddle batches

    const int m  = lane & 15;           // query row (A) / point col (B) owned by lane
    const int kb = (lane >> 4) * 2;     // K pair: lanes 0-15 -> K{0,1}, 16-31 -> K{2,3}

    // ---------------- A operand: queries ----------------
    const float* qp = new_xyz + ((size_t)b * PP + (size_t)(pbase + m)) * 3;
    const float qx = qp[0], qy = qp[1], qz = qp[2];
    const float q2 = qx * qx + qy * qy + qz * qz;
    // A(m,k) = { -2qx, -2qy, -2qz, |q|^2 }
    v2f A;
    if (kb == 0) { A[0] = -2.0f * qx; A[1] = -2.0f * qy; }
    else         { A[0] = -2.0f * qz; A[1] = q2;         }

    // lanes >= 16 are "always done" for the selection phase
    int cnt = (lane < 16) ? 0 : NSAMPLE;

    const float* xb = xyz + (size_t)b * NN * 3;

    // ---------------- scan N in 64-point chunks (4 WMMA tiles) ----------------
    for (int t = 0; t < NN / 64; ++t) {
        const int base = t * 64;

        // Issue all 4 sub-tile point loads up front (independent b96 loads)
        v2f   Bv[4];
        float thr[4];
#pragma unroll
        for (int u = 0; u < 4; ++u) {
            const float* pp = xb + (size_t)(base + u * 16 + m) * 3;
            const float px = pp[0], py = pp[1], pz = pp[2];
            const float p2 = px * px + py * py + pz * pz;
            thr[u] = RADIUS2 - p2;     // fold |p|^2 into the compare threshold
            if (kb == 0) { Bv[u][0] = px; Bv[u][1] = py;   }
            else         { Bv[u][0] = pz; Bv[u][1] = 1.0f; }
        }

        // Prefetch next chunk's points (speculative, dropped if OOB translation)
        if (base + 64 < NN) {
            __builtin_prefetch(xb + (size_t)(base + 64 + m) * 3, 0, 0);
            __builtin_prefetch(xb + (size_t)(base + 96 + m) * 3, 0, 0);
        }

        // 4 WMMAs: D = (-2q).p + |q|^2, accumulating into inline C = 0
        unsigned mrow[4];
#pragma unroll
        for (int u = 0; u < 4; ++u) {
            v8f Cz = {};
            v8f D = __builtin_amdgcn_wmma_f32_16x16x4_f32(
                false, A, false, Bv[u], (short)0, Cz, false, false);

            // in-ball test: D < r^2 - |p|^2 (per-lane threshold)
            const unsigned b0 = __builtin_amdgcn_ballot_w32(D[0] < thr[u]);
            const unsigned b1 = __builtin_amdgcn_ballot_w32(D[1] < thr[u]);
            const unsigned b2 = __builtin_amdgcn_ballot_w32(D[2] < thr[u]);
            const unsigned b3 = __builtin_amdgcn_ballot_w32(D[3] < thr[u]);
            const unsigned b4 = __builtin_amdgcn_ballot_w32(D[4] < thr[u]);
            const unsigned b5 = __builtin_amdgcn_ballot_w32(D[5] < thr[u]);
            const unsigned b6 = __builtin_amdgcn_ballot_w32(D[6] < thr[u]);
            const unsigned b7 = __builtin_amdgcn_ballot_w32(D[7] < thr[u]);

            // Route row-mask m to the owning lane (bits[15:0]=row r, [31:16]=row r+8)
            const unsigned u0 = (m & 1) ? b1 : b0;
            const unsigned u1 = (m & 1) ? b3 : b2;
            const unsigned u2 = (m & 1) ? b5 : b4;
            const unsigned u3 = (m & 1) ? b7 : b6;
            const unsigned w0 = (m & 2) ? u1 : u0;
            const unsigned w1 = (m & 2) ? u3 : u2;
            const unsigned w  = (m & 4) ? w1 : w0;
            mrow[u] = (m & 8) ? (w >> 16) : (w & 0xFFFFu);
        }

        // Concatenate pairs of 16-bit row masks -> 32 consecutive points per mask
        unsigned mm0 = mrow[0] | (mrow[1] << 16);
        unsigned mm1 = mrow[2] | (mrow[3] << 16);

        if (lane < 16) {
            while (mm0 && cnt < NSAMPLE) {          // points [base, base+32)
                const int bit = __builtin_ctz(mm0);
                mm0 &= mm0 - 1;
                s_idx[wave][m][cnt++] = base + bit;
            }
            while (mm1 && cnt < NSAMPLE) {          // points [base+32, base+64)
                const int bit = __builtin_ctz(mm1);
                mm1 &= mm1 - 1;
                s_idx[wave][m][cnt++] = base + 32 + bit;
            }
        }

        // wave-uniform early exit; EXEC is all-1s at every WMMA
        if (__builtin_amdgcn_ballot_w32(cnt >= NSAMPLE) == 0xFFFFFFFFu)
            break;
    }

    // ---------------- pad slots (first found, or -1 = empty) ----------------
    if (lane < 16) {
        const int first = (cnt > 0) ? s_idx[wave][m][0] : -1;
        for (int s = cnt; s < NSAMPLE; ++s) s_idx[wave][m][s] = first;
    }
    __syncthreads();

    // ---------------- fused grouping / gather / write ----------------
    const float* fb = features + (size_t)b * CC * NN;
    float* ob = out + (size_t)b * (3 + CC) * PP * NSAMPLE;
    const int s = lane;  // slot

    for (int q = 0; q < 16; ++q) {
        const int p = pbase + q;
        const int j = s_idx[wave][q][s];
        const size_t obase = (size_t)p * NSAMPLE + (size_t)s;

        if (j < 0) {
            // ball query found nothing -> all channels zero
#pragma unroll 4
            for (int ch = 0; ch < 3 + CC; ++ch)
                ob[(size_t)ch * ((size_t)PP * NSAMPLE) + obase] = 0.0f;
        } else {
            const float* pq = new_xyz + ((size_t)b * PP + (size_t)p) * 3;
            const float* pj = xb + (size_t)j * 3;
            ob[0 * ((size_t)PP * NSAMPLE) + obase] = pj[0] - pq[0];
            ob[1 * ((size_t)PP * NSAMPLE) + obase] = pj[1] - pq[1];
            ob[2 * ((size_t)PP * NSAMPLE) + obase] = pj[2] - pq[2];
#pragma unroll 4
            for (int c = 0; c < CC; ++c)
                ob[(size_t)(3 + c) * ((size_t)PP * NSAMPLE) + obase] =
                    fb[(size_t)c * NN + (size_t)j];
        }
    }
}

extern "C" void kernel_launch(void* const* d_in, const int* in_sizes, int n_in,
                              void* d_out, int out_size, void* d_ws, size_t ws_size,
                              hipStream_t stream) {
    (void)in_sizes; (void)n_in; (void)d_ws; (void)ws_size; (void)out_size;
    const float* xyz      = (const float*)d_in[0];  // (B, N, 3)
    const float* new_xyz  = (const float*)d_in[1];  // (B, P, 3)
    const float* features = (const float*)d_in[2];  // (B, C, N)
    float* out = (float*)d_out;                     // (B, 67, P, 32)

    const int tiles  = (BB * PP) / 16;              // 512 sixteen-query tiles
    const int blocks = tiles / WAVES_PER_BLOCK;     // 64 blocks of 256 threads
    qg_wmma_kernel<<<blocks, THREADS, 0, stream>>>(xyz, new_xyz, features, out);
}